// TailSAGE_SP_10866267259414
// MI455X (gfx1250) — compile-verified
//
#include <hip/hip_runtime.h>
#include <hip/hip_bf16.h>
#include <math.h>

// ---------------------------------------------------------------------------
// TailSAGE on gfx1250: f32 WMMA (V_WMMA_F32_16X16X4_F32) for all dense GEMMs,
// wave-per-edge SpMM with global_atomic_add_f32, per-row reductions for norms.
// GEMMs are N-register-blocked (one wave owns a 16 x (16*NT) strip; the A
// fragment is loaded once per K-step and reused for NT WMMAs) and software-
// pipelined one K-step deep so loads of step k+4 overlap WMMAs of step k.
// ---------------------------------------------------------------------------

typedef __attribute__((ext_vector_type(2))) float v2f;
typedef __attribute__((ext_vector_type(8))) float v8f;

#define NF 256   // feature width of all 256-wide stages

// -------------------------------- fill ------------------------------------
__global__ void fill0_kernel(float* __restrict__ p, size_t n) {
    size_t i = (size_t)blockIdx.x * blockDim.x + threadIdx.x;
    size_t stride = (size_t)gridDim.x * blockDim.x;
    for (; i < n; i += stride) p[i] = 0.0f;
}

// -------------------------------- SpMM ------------------------------------
// neighbor[row[e], :] += edge_weight[e] * x[col[e], :]
// One wave (32 lanes) per edge; lane handles 8 consecutive floats (F=256).
__global__ __launch_bounds__(256)
void spmm_kernel(const float* __restrict__ x,
                 const int* __restrict__ col,
                 const int* __restrict__ row,
                 const float* __restrict__ ew,
                 float* __restrict__ out, int E) {
    int wid  = (int)((blockIdx.x * blockDim.x + threadIdx.x) >> 5);
    int lane = threadIdx.x & 31;
    if (wid >= E) return;
    int   src = col[wid];
    int   dst = row[wid];
    float w   = ew[wid];
    const float4* xs = (const float4*)(x + (size_t)src * NF);
    float* o = out + (size_t)dst * NF + lane * 8;
    float4 v0 = xs[lane * 2 + 0];
    float4 v1 = xs[lane * 2 + 1];
    unsafeAtomicAdd(o + 0, v0.x * w);
    unsafeAtomicAdd(o + 1, v0.y * w);
    unsafeAtomicAdd(o + 2, v0.z * w);
    unsafeAtomicAdd(o + 3, v0.w * w);
    unsafeAtomicAdd(o + 4, v1.x * w);
    unsafeAtomicAdd(o + 5, v1.y * w);
    unsafeAtomicAdd(o + 6, v1.z * w);
    unsafeAtomicAdd(o + 7, v1.w * w);
}

// ----------------------------- WMMA GEMM ----------------------------------
// C[:, colOff:colOff+Nc] = A1 @ W1^T (+ A2 @ W2^T) (+ bias)
// A row-major [M,K] (ld=ldA), W row-major [Nc,K] (ld=K), C ld=ldC.
// One wave per 16 x (16*NT) output strip; K consumed 4 at a time with
// V_WMMA_F32_16X16X4_F32.  ISA fragment layout (wave32):
//   A 16x4 : lane l<16 holds {A[m0+l,k+0], A[m0+l,k+1]}, l>=16 -> k+2,k+3
//   B 4x16 : lane l<16 holds {W[n0+l,k+0], W[n0+l,k+1]}, l>=16 -> k+2,k+3
//   C/D    : vgpr v, lane l<16 -> C[m0+v, n0+l]; l>=16 -> C[m0+v+8, n0+l-16]
// GUARD=false requires Nc to be a multiple of 16*NT (mask-free inner loop).
template <int NT, bool GUARD>
struct Frag {
    v2f a;
    v2f b[NT];
    __device__ __forceinline__
    void load(const float* __restrict__ arow, const float* const wrow[NT],
              const float wmask[NT], int k) {
        a = *(const v2f*)(arow + k);
#pragma unroll
        for (int j = 0; j < NT; ++j) {
            v2f t = *(const v2f*)(wrow[j] + k);
            if (GUARD) { t.x *= wmask[j]; t.y *= wmask[j]; }
            b[j] = t;
        }
    }
};

template <int NT, bool GUARD>
__global__ __launch_bounds__(256)
void gemm_wmma_kernel(const float* __restrict__ A1, const float* __restrict__ W1,
                      const float* __restrict__ A2, const float* __restrict__ W2,
                      float* __restrict__ Cd, const float* __restrict__ bias,
                      int M, int K, int Nc, int ldA, int ldC, int colOff) {
    int lane = threadIdx.x & 31;
    int wv   = threadIdx.x >> 5;
    int tiles_n = (Nc + 15) >> 4;
    int groups  = (tiles_n + NT - 1) / NT;
    int total   = (M >> 4) * groups;
    int t = blockIdx.x * (blockDim.x >> 5) + wv;
    if (t >= total) return;                 // wave-uniform: EXEC stays all-1s

    int m0 = (t / groups) << 4;
    int n0 = (t % groups) * (NT * 16);
    int l  = lane & 15;
    int hl = lane >> 4;                     // 0: K+0..1, 1: K+2..3

    int   nn[NT];
    float wmask[NT];
    const float* wrow1[NT];
#pragma unroll
    for (int j = 0; j < NT; ++j) {
        int n = n0 + j * 16 + l;
        bool ok = GUARD ? (n < Nc) : true;
        nn[j]    = n;
        wmask[j] = ok ? 1.0f : 0.0f;
        wrow1[j] = W1 + (size_t)(ok ? n : 0) * K + 2 * hl;
    }
    const float* arow1 = A1 + (size_t)(m0 + l) * ldA + 2 * hl;

    v8f acc[NT];
#pragma unroll
    for (int j = 0; j < NT; ++j) acc[j] = (v8f){};

    // ---- pass 1: software-pipelined K loop (prefetch depth 1) ----
    {
        Frag<NT, GUARD> cur, nxt;
        cur.load(arow1, wrow1, wmask, 0);
        for (int k = 0; k < K - 4; k += 4) {
            nxt.load(arow1, wrow1, wmask, k + 4);
#pragma unroll
            for (int j = 0; j < NT; ++j)
                acc[j] = __builtin_amdgcn_wmma_f32_16x16x4_f32(
                    false, cur.a, false, cur.b[j], (short)0, acc[j], false, false);
            cur = nxt;
        }
#pragma unroll
        for (int j = 0; j < NT; ++j)
            acc[j] = __builtin_amdgcn_wmma_f32_16x16x4_f32(
                false, cur.a, false, cur.b[j], (short)0, acc[j], false, false);
    }

    // ---- pass 2 (fused second product) ----
    if (A2 != nullptr) {
        const float* arow2 = A2 + (size_t)(m0 + l) * ldA + 2 * hl;
        const float* wrow2[NT];
#pragma unroll
        for (int j = 0; j < NT; ++j)
            wrow2[j] = W2 + (size_t)(GUARD ? (wmask[j] != 0.0f ? nn[j] : 0) : nn[j]) * K + 2 * hl;

        Frag<NT, GUARD> cur, nxt;
        cur.load(arow2, wrow2, wmask, 0);
        for (int k = 0; k < K - 4; k += 4) {
            nxt.load(arow2, wrow2, wmask, k + 4);
#pragma unroll
            for (int j = 0; j < NT; ++j)
                acc[j] = __builtin_amdgcn_wmma_f32_16x16x4_f32(
                    false, cur.a, false, cur.b[j], (short)0, acc[j], false, false);
            cur = nxt;
        }
#pragma unroll
        for (int j = 0; j < NT; ++j)
            acc[j] = __builtin_amdgcn_wmma_f32_16x16x4_f32(
                false, cur.a, false, cur.b[j], (short)0, acc[j], false, false);
    }

    int mrow = m0 + hl * 8;
#pragma unroll
    for (int j = 0; j < NT; ++j) {
        if (!GUARD || nn[j] < Nc) {
            float bv = (bias != nullptr) ? bias[nn[j]] : 0.0f;
            float* cptr = Cd + (size_t)mrow * ldC + colOff + nn[j];
#pragma unroll
            for (int v = 0; v < 8; ++v)
                cptr[(size_t)v * ldC] = acc[j][v] + bv;
        }
    }
}

// --------------------------- relation update -------------------------------
// gamma = lrelu(G,0.2)+1 ; beta = lrelu(B,0.2)
// m = ft + gamma*r + beta - nb ; mnorm[row] = ||m||
// if (*head == 0) nb = nb + m/(norm+1)
__global__ __launch_bounds__(256)
void relation_update_kernel(const float* __restrict__ ft,
                            const float* __restrict__ G,
                            const float* __restrict__ Bt,
                            const float* __restrict__ r,
                            const float* __restrict__ norm,
                            const int* __restrict__ head,
                            float* __restrict__ nb,
                            float* __restrict__ mnorm) {
    int rowi = blockIdx.x;
    int c = threadIdx.x;
    size_t idx = (size_t)rowi * NF + c;

    float nbv = nb[idx];
    float g = G[idx];
    float b = Bt[idx];
    g = (g > 0.0f ? g : 0.2f * g) + 1.0f;
    b = (b > 0.0f ? b : 0.2f * b);
    float m = ft[idx] + (g * r[c] + b) - nbv;

    float s = m * m;
#pragma unroll
    for (int off = 16; off > 0; off >>= 1) s += __shfl_xor(s, off);
    __shared__ float red[8];
    int wid = threadIdx.x >> 5, ln = threadIdx.x & 31;
    if (ln == 0) red[wid] = s;
    __syncthreads();
    if (threadIdx.x == 0) {
        float tt = 0.0f;
#pragma unroll
        for (int i = 0; i < 8; ++i) tt += red[i];
        mnorm[rowi] = sqrtf(tt);
    }
    if (*head == 0) nb[idx] = nbv + m / (norm[rowi] + 1.0f);
}

// ------------------------------ elu + l2norm -------------------------------
__global__ __launch_bounds__(256)
void elu_l2norm_kernel(float* __restrict__ X, int D) {
    int rowi = blockIdx.x;
    int c = threadIdx.x;
    bool act = c < D;
    size_t idx = (size_t)rowi * D + c;
    float v = 0.0f;
    if (act) {
        v = X[idx];
        v = (v > 0.0f) ? v : (expf(v) - 1.0f);
    }
    float s = v * v;
#pragma unroll
    for (int off = 16; off > 0; off >>= 1) s += __shfl_xor(s, off);
    __shared__ float red[8];
    __shared__ float nrm;
    int wid = threadIdx.x >> 5, ln = threadIdx.x & 31;
    if (ln == 0) red[wid] = s;
    __syncthreads();
    if (threadIdx.x == 0) {
        float tt = 0.0f;
#pragma unroll
        for (int i = 0; i < 8; ++i) tt += red[i];
        nrm = sqrtf(tt);
    }
    __syncthreads();
    if (act) X[idx] = v / fmaxf(nrm, 1e-12f);
}

// ------------------------------ launcher -----------------------------------
static inline void launch_gemm(const float* A1, const float* W1,
                               const float* A2, const float* W2,
                               float* Cd, const float* bias,
                               int M, int K, int Nc, int ldA, int ldC, int colOff,
                               hipStream_t stream) {
    int tiles_n = (Nc + 15) >> 4;
    if ((Nc & 63) == 0) {
        int groups = tiles_n / 4;
        int total  = (M >> 4) * groups;
        int blocks = (total + 7) / 8;        // 8 waves / 256-thread block
        gemm_wmma_kernel<4, false><<<blocks, 256, 0, stream>>>(
            A1, W1, A2, W2, Cd, bias, M, K, Nc, ldA, ldC, colOff);
    } else {
        int groups = (tiles_n + 2) / 3;
        int total  = (M >> 4) * groups;
        int blocks = (total + 7) / 8;
        gemm_wmma_kernel<3, true><<<blocks, 256, 0, stream>>>(
            A1, W1, A2, W2, Cd, bias, M, K, Nc, ldA, ldC, colOff);
    }
}

extern "C" void kernel_launch(void* const* d_in, const int* in_sizes, int n_in,
                              void* d_out, int out_size, void* d_ws, size_t ws_size,
                              hipStream_t stream) {
    const float* x    = (const float*)d_in[0];   // [N,256]
    const int*   rowi = (const int*)  d_in[1];   // [E]
    const int*   coli = (const int*)  d_in[2];   // [E]
    const float* ew   = (const float*)d_in[3];   // [E]
    const float* norm = (const float*)d_in[4];   // [N,1]
    const int*   head = (const int*)  d_in[5];   // scalar
    const float* w1   = (const float*)d_in[6];   // [128,256]
    const float* w2   = (const float*)d_in[7];   // [40,256]
    const float* fcw  = (const float*)d_in[8];   // [40,80]
    const float* fcb  = (const float*)d_in[9];   // [40]
    const float* r1g1 = (const float*)d_in[10];
    const float* r1g2 = (const float*)d_in[11];
    const float* r1b1 = (const float*)d_in[12];
    const float* r1b2 = (const float*)d_in[13];
    const float* r1r  = (const float*)d_in[14];
    const float* r2g1 = (const float*)d_in[15];
    const float* r2g2 = (const float*)d_in[16];
    const float* r2b1 = (const float*)d_in[17];
    const float* r2b2 = (const float*)d_in[18];
    const float* r2r  = (const float*)d_in[19];

    const int N = in_sizes[4];          // norm is [N,1]
    const int E = in_sizes[1];
    const int C = in_sizes[9];          // fc_b is [C] = 40

    float* out   = (float*)d_out;
    float* m1n   = out + (size_t)N * C;        // [N]
    float* m2n   = m1n + N;                    // [N]

    float* ws   = (float*)d_ws;
    size_t pane = (size_t)N * NF;
    float* buf0 = ws;                 // neighbor / neighbor1   [N,256]
    float* buf1 = ws + pane;          // gamma-pre, later x2pre [N,256]
    float* buf2 = ws + 2 * pane;      // beta-pre               [N,256]
    float* buf3 = ws + 3 * pane;      // x1                     [N,256]

    int spmmBlocks = (E + 7) / 8;     // one wave per edge, 8 waves/block

    // ---- layer 1 ----
    fill0_kernel<<<4096, 256, 0, stream>>>(buf0, pane);
    spmm_kernel<<<spmmBlocks, 256, 0, stream>>>(x, coli, rowi, ew, buf0, E);

    launch_gemm(x, r1g1, buf0, r1g2, buf1, nullptr, N, NF, NF, NF, NF, 0, stream);
    launch_gemm(x, r1b1, buf0, r1b2, buf2, nullptr, N, NF, NF, NF, NF, 0, stream);
    relation_update_kernel<<<N, 256, 0, stream>>>(x, buf1, buf2, r1r, norm, head,
                                                  buf0, m1n);

    launch_gemm(x,    w1, nullptr, nullptr, buf3, nullptr, N, NF, 128, NF, NF, 0,   stream);
    launch_gemm(buf0, w1, nullptr, nullptr, buf3, nullptr, N, NF, 128, NF, NF, 128, stream);
    elu_l2norm_kernel<<<N, 256, 0, stream>>>(buf3, 256);

    // ---- layer 2 ----
    fill0_kernel<<<4096, 256, 0, stream>>>(buf0, pane);
    spmm_kernel<<<spmmBlocks, 256, 0, stream>>>(buf3, coli, rowi, ew, buf0, E);

    launch_gemm(buf3, r2g1, buf0, r2g2, buf1, nullptr, N, NF, NF, NF, NF, 0, stream);
    launch_gemm(buf3, r2b1, buf0, r2b2, buf2, nullptr, N, NF, NF, NF, NF, 0, stream);
    relation_update_kernel<<<N, 256, 0, stream>>>(buf3, buf1, buf2, r2r, norm, head,
                                                  buf0, m2n);

    // x2pre = concat(x1@w2^T, nb1@w2^T)  -> buf1 [N,80]
    launch_gemm(buf3, w2, nullptr, nullptr, buf1, nullptr, N, NF, C, NF, 2 * C, 0, stream);
    launch_gemm(buf0, w2, nullptr, nullptr, buf1, nullptr, N, NF, C, NF, 2 * C, C, stream);
    elu_l2norm_kernel<<<N, 256, 0, stream>>>(buf1, 2 * C);

    // out = x2 @ fc_w^T + fc_b  [N,40]
    launch_gemm(buf1, fcw, nullptr, nullptr, out, fcb, N, 2 * C, C, 2 * C, C, 0, stream);
}